// GAT_541165879571
// MI455X (gfx1250) — compile-verified
//
#include <hip/hip_runtime.h>
#include <math.h>

// ---- problem constants (match reference) ----
#define NN      50000
#define EE      800000
#define EP      (EE + NN)      // edges + self loops
#define HID     64
#define HEADS   4
#define F1      (HEADS * HID)  // 256
#define NCLS    16
#define SLOPE   0.2f

typedef __attribute__((ext_vector_type(2))) float v2f;
typedef __attribute__((ext_vector_type(8))) float v8f;

// ---------------------------------------------------------------------------
// Generic f32 WMMA GEMM: C[M x N] = A[M x K] * B[K x N], row-major.
// One wave computes one 16x16 tile using V_WMMA_F32_16X16X4_F32, K stepped by 4.
// Requires M%16==0, N%16==0, K%4==0 (true here: M=50000, N=256/16, K=64/256).
// ---------------------------------------------------------------------------
__global__ __launch_bounds__(128) void wmma_gemm_f32(
    const float* __restrict__ A, const float* __restrict__ B,
    float* __restrict__ C, int M, int N, int K) {
  const int wavesPerBlock = blockDim.x >> 5;
  const int wave = blockIdx.x * wavesPerBlock + (threadIdx.x >> 5);
  const int lane = threadIdx.x & 31;
  const int tilesN = N >> 4;
  const int tilesM = M >> 4;
  if (wave >= tilesM * tilesN) return;   // wave-uniform guard (EXEC all-1 inside)
  const int tm = wave / tilesN;
  const int tn = wave - tm * tilesN;
  const int m0 = tm << 4, n0 = tn << 4;

  // A 16x4 layout: lanes 0-15 hold M=lane,K={0,1}; lanes 16-31 hold M=lane-16,K={2,3}
  // B 4x16 layout: lanes 0-15 hold N=lane,K={0,1}; lanes 16-31 hold N=lane-16,K={2,3}
  const int r  = lane & 15;
  const int kq = (lane >> 4) << 1;   // 0 or 2

  const float* Ap = A + (size_t)(m0 + r) * K + kq;
  const float* Bp = B + (size_t)kq * N + (n0 + r);

  v8f c = {0.f, 0.f, 0.f, 0.f, 0.f, 0.f, 0.f, 0.f};
  for (int k = 0; k < K; k += 4) {
    v2f a, b;
    a.x = Ap[k];
    a.y = Ap[k + 1];
    b.x = Bp[(size_t)k * N];
    b.y = Bp[(size_t)(k + 1) * N];
    c = __builtin_amdgcn_wmma_f32_16x16x4_f32(
        /*neg_a=*/false, a, /*neg_b=*/false, b,
        /*c_mod=*/(short)0, c, /*reuse_a=*/false, /*reuse_b=*/false);
  }

  // C/D layout: lanes 0-15: N=lane, VGPR v -> M=v ; lanes 16-31: N=lane-16, M=v+8
  const int crow = m0 + ((lane >> 4) << 3);
  const int ccol = n0 + r;
  #pragma unroll
  for (int v = 0; v < 8; v++)
    C[(size_t)(crow + v) * N + ccol] = c[v];
}

// ---------------------------------------------------------------------------
__global__ void fill_f32(float* p, float v, int n) {
  int i = blockIdx.x * blockDim.x + threadIdx.x;
  if (i < n) p[i] = v;
}

// a_src[n,h] = dot(xh[n,h,:], att_s[h,:]) ; same for a_dst
__global__ void att_scores(const float* __restrict__ xh,
                           const float* __restrict__ att_s,
                           const float* __restrict__ att_d,
                           float* __restrict__ aS, float* __restrict__ aD,
                           int n_nodes, int H, int D) {
  int i = blockIdx.x * blockDim.x + threadIdx.x;
  if (i >= n_nodes * H) return;
  int n = i / H, h = i - n * H;
  const float* row = xh + (size_t)n * H * D + (size_t)h * D;
  const float* as = att_s + h * D;
  const float* ad = att_d + h * D;
  float s = 0.f, d = 0.f;
  for (int k = 0; k < D; k++) {
    float v = row[k];
    s = fmaf(v, as[k], s);
    d = fmaf(v, ad[k], d);
  }
  aS[i] = s;
  aD[i] = d;
}

__device__ __forceinline__ void edge_sd(const long long* ei, int e, int E_,
                                        long long& s, long long& d) {
  if (e < E_) { s = ei[e]; d = ei[(size_t)E_ + e]; }
  else        { s = d = (long long)(e - E_); }
}

// segment max of leaky_relu(a_src[src]+a_dst[dst]) over dst
__global__ void edge_max(const long long* __restrict__ ei,
                         const float* __restrict__ aS, const float* __restrict__ aD,
                         float* __restrict__ m, int E_, int n_nodes, int H) {
  int e = blockIdx.x * blockDim.x + threadIdx.x;
  if (e >= E_ + n_nodes) return;
  long long s, d;
  edge_sd(ei, e, E_, s, d);
  for (int h = 0; h < H; h++) {
    float v = aS[(size_t)s * H + h] + aD[(size_t)d * H + h];
    v = v > 0.f ? v : SLOPE * v;
    atomicMax(&m[(size_t)d * H + h], v);
  }
}

// ex = exp(e - m[dst]) ; denom[dst] += ex ; stash ex per edge
__global__ void edge_expsum(const long long* __restrict__ ei,
                            const float* __restrict__ aS, const float* __restrict__ aD,
                            const float* __restrict__ m,
                            float* __restrict__ ex, float* __restrict__ den,
                            int E_, int n_nodes, int H) {
  int e = blockIdx.x * blockDim.x + threadIdx.x;
  if (e >= E_ + n_nodes) return;
  long long s, d;
  edge_sd(ei, e, E_, s, d);
  for (int h = 0; h < H; h++) {
    float v = aS[(size_t)s * H + h] + aD[(size_t)d * H + h];
    v = v > 0.f ? v : SLOPE * v;
    float x = expf(v - m[(size_t)d * H + h]);
    ex[(size_t)e * H + h] = x;
    atomicAdd(&den[(size_t)d * H + h], x);
  }
}

// out[dst,:] += alpha(e,h) * xh[src,:]  (float4-vectorized; 4 feats share a head)
__global__ void edge_aggregate(const long long* __restrict__ ei,
                               const float* __restrict__ xh,
                               const float* __restrict__ ex,
                               const float* __restrict__ den,
                               float* __restrict__ out,
                               int E_, int n_nodes, int H, int D) {
  const int F = H * D, F4 = F >> 2;
  int idx = blockIdx.x * blockDim.x + threadIdx.x;
  int total = (E_ + n_nodes) * F4;
  if (idx >= total) return;
  int e = idx / F4;
  int j = idx - e * F4;
  long long s, d;
  edge_sd(ei, e, E_, s, d);
  int h = (j << 2) / D;
  float alpha = ex[(size_t)e * H + h] / (den[(size_t)d * H + h] + 1e-16f);
  const float4 v = ((const float4*)(xh + (size_t)s * F))[j];
  float* o = out + (size_t)d * F + (j << 2);
  atomicAdd(o + 0, alpha * v.x);
  atomicAdd(o + 1, alpha * v.y);
  atomicAdd(o + 2, alpha * v.z);
  atomicAdd(o + 3, alpha * v.w);
}

// in-place h = elu(acc + b)
__global__ void bias_elu(float* __restrict__ acc, const float* __restrict__ b,
                         int n_nodes, int F) {
  int i = blockIdx.x * blockDim.x + threadIdx.x;
  if (i >= n_nodes * F) return;
  float v = acc[i] + b[i % F];
  acc[i] = v > 0.f ? v : (expf(v) - 1.f);
}

// in-place per-row: out = log_softmax(out + b)
__global__ void logsoftmax_bias(float* __restrict__ out, const float* __restrict__ b,
                                int n_nodes, int C) {
  int n = blockIdx.x * blockDim.x + threadIdx.x;
  if (n >= n_nodes) return;
  float v[NCLS];
  float mx = -3.4e38f;
  for (int c = 0; c < C; c++) {
    v[c] = out[(size_t)n * C + c] + b[c];
    mx = fmaxf(mx, v[c]);
  }
  float sum = 0.f;
  for (int c = 0; c < C; c++) sum += expf(v[c] - mx);
  float lse = logf(sum) + mx;
  for (int c = 0; c < C; c++) out[(size_t)n * C + c] = v[c] - lse;
}

// ---------------------------------------------------------------------------
extern "C" void kernel_launch(void* const* d_in, const int* in_sizes, int n_in,
                              void* d_out, int out_size, void* d_ws, size_t ws_size,
                              hipStream_t stream) {
  const float*     x     = (const float*)d_in[0];
  const long long* ei    = (const long long*)d_in[1];  // int64 [2,E]
  const float*     W1    = (const float*)d_in[2];
  const float*     attS1 = (const float*)d_in[3];
  const float*     attD1 = (const float*)d_in[4];
  const float*     b1    = (const float*)d_in[5];
  const float*     W2    = (const float*)d_in[6];
  const float*     attS2 = (const float*)d_in[7];
  const float*     attD2 = (const float*)d_in[8];
  const float*     b2    = (const float*)d_in[9];
  float* out = (float*)d_out;

  // workspace layout (floats)
  float* ws   = (float*)d_ws;
  float* xh1  = ws;                       // N*F1  = 12,800,000
  float* acc1 = xh1  + (size_t)NN * F1;   // N*F1  (becomes h after bias_elu)
  float* aS1  = acc1 + (size_t)NN * F1;   // N*H
  float* aD1  = aS1  + (size_t)NN * HEADS;
  float* m1   = aD1  + (size_t)NN * HEADS;
  float* den1 = m1   + (size_t)NN * HEADS;
  float* ex1  = den1 + (size_t)NN * HEADS;          // EP*H = 3,400,000
  float* xh2  = ex1  + (size_t)EP * HEADS;          // N*C
  float* aS2  = xh2  + (size_t)NN * NCLS;
  float* aD2  = aS2  + (size_t)NN;
  float* m2   = aD2  + (size_t)NN;
  float* den2 = m2   + (size_t)NN;
  float* ex2  = den2 + (size_t)NN;                  // EP

  const int T = 256;
  auto cdiv = [](int a, int b) { return (a + b - 1) / b; };

  // ---- init accumulators / softmax state ----
  fill_f32<<<cdiv(NN * F1, T), T, 0, stream>>>(acc1, 0.f, NN * F1);
  fill_f32<<<cdiv(NN * HEADS, T), T, 0, stream>>>(m1, -3.0e38f, NN * HEADS);
  fill_f32<<<cdiv(NN * HEADS, T), T, 0, stream>>>(den1, 0.f, NN * HEADS);
  fill_f32<<<cdiv(NN * NCLS, T), T, 0, stream>>>(out, 0.f, NN * NCLS);
  fill_f32<<<cdiv(NN, T), T, 0, stream>>>(m2, -3.0e38f, NN);
  fill_f32<<<cdiv(NN, T), T, 0, stream>>>(den2, 0.f, NN);

  // ---- conv1 ----
  {
    int tiles = (NN / 16) * (F1 / 16);          // 50000 waves
    wmma_gemm_f32<<<cdiv(tiles, 4), 128, 0, stream>>>(x, W1, xh1, NN, F1, HID);
  }
  att_scores<<<cdiv(NN * HEADS, T), T, 0, stream>>>(xh1, attS1, attD1, aS1, aD1,
                                                    NN, HEADS, HID);
  edge_max<<<cdiv(EP, T), T, 0, stream>>>(ei, aS1, aD1, m1, EE, NN, HEADS);
  edge_expsum<<<cdiv(EP, T), T, 0, stream>>>(ei, aS1, aD1, m1, ex1, den1, EE, NN, HEADS);
  edge_aggregate<<<cdiv(EP * (F1 / 4), T), T, 0, stream>>>(ei, xh1, ex1, den1, acc1,
                                                           EE, NN, HEADS, HID);
  bias_elu<<<cdiv(NN * F1, T), T, 0, stream>>>(acc1, b1, NN, F1);

  // ---- conv2 ----
  {
    int tiles = (NN / 16) * (NCLS / 16);        // 3125 waves
    wmma_gemm_f32<<<cdiv(tiles, 4), 128, 0, stream>>>(acc1, W2, xh2, NN, NCLS, F1);
  }
  att_scores<<<cdiv(NN, T), T, 0, stream>>>(xh2, attS2, attD2, aS2, aD2, NN, 1, NCLS);
  edge_max<<<cdiv(EP, T), T, 0, stream>>>(ei, aS2, aD2, m2, EE, NN, 1);
  edge_expsum<<<cdiv(EP, T), T, 0, stream>>>(ei, aS2, aD2, m2, ex2, den2, EE, NN, 1);
  edge_aggregate<<<cdiv(EP * (NCLS / 4), T), T, 0, stream>>>(ei, xh2, ex2, den2, out,
                                                             EE, NN, 1, NCLS);
  logsoftmax_bias<<<cdiv(NN, T), T, 0, stream>>>(out, b2, NN, NCLS);
}